// Fingerprint__26731876450917
// MI455X (gfx1250) — compile-verified
//
#include <hip/hip_runtime.h>

#define BN 256
#define LN 128
#define DN 6
#define AFD 39
#define BFD 10
#define FP 200
#define NROWS (BN * LN)        // 32768
#define NDROWS (NROWS * DN)    // 196608
#define NT_COLS 13             // ceil(200/16)
#define NEGBIG -9e8f
#define WPAD 204               // padded LDS row stride (dwords), conflict-free & 16B aligned

typedef float v2f __attribute__((ext_vector_type(2)));
typedef float v8f __attribute__((ext_vector_type(8)));
typedef unsigned int v4u __attribute__((ext_vector_type(4)));
typedef int v8i_t __attribute__((ext_vector_type(8)));
typedef int v4i_t __attribute__((ext_vector_type(4)));

__device__ __forceinline__ float lrelu(float x) { return x > 0.f ? x : 0.01f * x; }
__device__ __forceinline__ float elu1(float x)  { return x > 0.f ? x : (__expf(x) - 1.f); }
__device__ __forceinline__ float sigmoidf(float x) { return 1.f / (1.f + __expf(-x)); }

// ---------------------------------------------------------------------------
// Kernel 1: Y[M,FP] = leaky_relu(X[M,K] @ W[FP,K]^T + bias)   (fp32 WMMA)
// ---------------------------------------------------------------------------
__global__ void fc_lrelu_kernel(const float* __restrict__ X,
                                const float* __restrict__ W,
                                const float* __restrict__ bias,
                                float* __restrict__ Y,
                                int M, int K) {
  int wave = blockIdx.x * (blockDim.x >> 5) + (threadIdx.x >> 5);
  int total = (M >> 4) * NT_COLS;
  if (wave >= total) return;
  int rb = (wave / NT_COLS) << 4;
  int cb = (wave % NT_COLS) << 4;
  int lane = threadIdx.x & 31, half = lane >> 4, lm = lane & 15;
  int n = cb + lm;
  int nc = (n < FP) ? n : (FP - 1);
  const float* xr = X + (size_t)(rb + lm) * K;
  const float* wr = W + (size_t)nc * K;
  v8f acc = {};
  int Kp = (K + 3) & ~3;
  for (int k0 = 0; k0 < Kp; k0 += 4) {
    int ka = k0 + 2 * half;
    float a0 = xr[(ka     < K) ? ka     : 0];
    float a1 = xr[(ka + 1 < K) ? ka + 1 : 0];
    float b0 = wr[(ka     < K) ? ka     : 0];
    float b1 = wr[(ka + 1 < K) ? ka + 1 : 0];
    v2f a, b;
    a.x = (ka < K) ? a0 : 0.f;  a.y = (ka + 1 < K) ? a1 : 0.f;
    b.x = (ka < K) ? b0 : 0.f;  b.y = (ka + 1 < K) ? b1 : 0.f;
    acc = __builtin_amdgcn_wmma_f32_16x16x4_f32(false, a, false, b, (short)0, acc, false, false);
  }
  if (n >= FP) return;
  float bv = bias[n];
#pragma unroll
  for (int v = 0; v < 8; ++v) {
    int m = rb + v + 8 * half;
    Y[(size_t)m * FP + n] = lrelu(acc[v] + bv);
  }
}

// ---------------------------------------------------------------------------
// Kernel 2: neighbor_fc — A rows gathered as concat(atom_nb, bond_nb), K=49
// ---------------------------------------------------------------------------
__global__ void neighbor_fc_kernel(const float* __restrict__ atom_list,
                                   const float* __restrict__ bond_list,
                                   const int* __restrict__ adl,
                                   const int* __restrict__ bdl,
                                   const float* __restrict__ W,
                                   const float* __restrict__ bias,
                                   float* __restrict__ nb) {
  const int K = AFD + BFD;  // 49
  int wave = blockIdx.x * (blockDim.x >> 5) + (threadIdx.x >> 5);
  int total = (NDROWS >> 4) * NT_COLS;
  if (wave >= total) return;
  int rb = (wave / NT_COLS) << 4;
  int cb = (wave % NT_COLS) << 4;
  int lane = threadIdx.x & 31, half = lane >> 4, lm = lane & 15;
  int n = cb + lm;
  int nc = (n < FP) ? n : (FP - 1);
  int r = rb + lm;                     // row in [0, N*D)
  int bidx = r / (LN * DN);
  int a_idx = adl[r];
  int b_idx = bdl[r];
  const float* arow = atom_list + ((size_t)bidx * LN + a_idx) * AFD;
  const float* brow = bond_list + ((size_t)bidx * LN + b_idx) * BFD;
  const float* wr = W + (size_t)nc * K;
  v8f acc = {};
  for (int k0 = 0; k0 < 52; k0 += 4) {   // K padded to 52
    int ka = k0 + 2 * half;
    v2f a, b;
#pragma unroll
    for (int j = 0; j < 2; ++j) {
      int k = ka + j;
      float va = arow[(k < AFD) ? k : 0];
      float vb = brow[(k >= AFD && k < K) ? (k - AFD) : 0];
      float e = (k < AFD) ? va : ((k < K) ? vb : 0.f);
      float w = (k < K) ? wr[k] : 0.f;
      if (j == 0) { a.x = e; b.x = w; } else { a.y = e; b.y = w; }
    }
    acc = __builtin_amdgcn_wmma_f32_16x16x4_f32(false, a, false, b, (short)0, acc, false, false);
  }
  if (n >= FP) return;
  float bv = bias[n];
#pragma unroll
  for (int v = 0; v < 8; ++v) {
    int m = rb + v + 8 * half;
    nb[(size_t)m * FP + n] = lrelu(acc[v] + bv);
  }
}

// ---------------------------------------------------------------------------
// Kernel 3a: per-(b,l) attention scores + softmax over D, fold weights:
//   wnb[r,:] = sum_d aw[d]*nbrow_d[:],  s_aw[r] = sum_d aw[d]
// ---------------------------------------------------------------------------
__global__ void attn_score_kernel(const float* __restrict__ cur,
                                  const float* __restrict__ nbsrc,
                                  int gather_mode,
                                  const int* __restrict__ adl,
                                  const float* __restrict__ aw_w,   // [2*FP]
                                  const float* __restrict__ aw_b,   // [1]
                                  float* __restrict__ wnb,
                                  float* __restrict__ s_aw) {
  int r = blockIdx.x * (blockDim.x >> 5) + (threadIdx.x >> 5);
  if (r >= NROWS) return;
  int lane = threadIdx.x & 31;
  int bidx = r / LN;
  const float* c = cur + (size_t)r * FP;
  const float* nrows[DN];
  float sc[DN], msk[DN], aw[DN];
  float ab = aw_b[0];
#pragma unroll
  for (int d = 0; d < DN; ++d) {
    int ai = adl[(size_t)r * DN + d];
    const float* nr = gather_mode ? nbsrc + ((size_t)bidx * LN + ai) * FP
                                  : nbsrc + ((size_t)r * DN + d) * FP;
    nrows[d] = nr;
    float p = 0.f;
    for (int k = lane; k < FP; k += 32)
      p += c[k] * aw_w[k] + nr[k] * aw_w[FP + k];
    for (int off = 16; off > 0; off >>= 1) p += __shfl_xor(p, off, 32);
    bool pad = (ai == LN - 1);
    msk[d] = pad ? 0.f : 1.f;
    sc[d] = lrelu(p + ab) + (pad ? NEGBIG : 0.f);
  }
  float mx = sc[0];
#pragma unroll
  for (int d = 1; d < DN; ++d) mx = fmaxf(mx, sc[d]);
  float se = 0.f;
#pragma unroll
  for (int d = 0; d < DN; ++d) { aw[d] = __expf(sc[d] - mx); se += aw[d]; }
  float inv = 1.f / se, ssum = 0.f;
#pragma unroll
  for (int d = 0; d < DN; ++d) { aw[d] = aw[d] * inv * msk[d]; ssum += aw[d]; }
  if (lane == 0) s_aw[r] = ssum;
  for (int k = lane; k < FP; k += 32) {
    float acc = 0.f;
#pragma unroll
    for (int d = 0; d < DN; ++d) acc += aw[d] * nrows[d][k];
    wnb[(size_t)r * FP + k] = acc;
  }
}

// ---------------------------------------------------------------------------
// Kernel 3b: ctx = elu(X[M,FP] @ W[FP,FP]^T + s_aw[m]*bias[n])
// Block = 4 row-tiles x 1 col-tile. W tile (16 rows x 200) staged into LDS by
// the Tensor Data Mover (TDM), B fragments read from LDS.
// ---------------------------------------------------------------------------
__global__ void attend_elu_kernel(const float* __restrict__ X,
                                  const float* __restrict__ W,
                                  const float* __restrict__ bias,
                                  const float* __restrict__ srow,
                                  float* __restrict__ Y, int M) {
  extern __shared__ float sW[];  // [16][FP] = 12800 B
  int cb = (blockIdx.x % NT_COLS) << 4;
  int rg = (blockIdx.x / NT_COLS) << 2;  // row-tile group (4 tiles)
  int lane = threadIdx.x & 31, half = lane >> 4, lm = lane & 15;

  if ((threadIdx.x >> 5) == 0) {
    // Build 2-D TDM descriptor: copy W[cb .. cb+15][0..199] -> LDS.
    // tensor_dim1 = remaining rows from tile start so edge tile gets OOB zeros.
    unsigned lds0 = (unsigned)(uintptr_t)&sW[0];
    unsigned long long ga = (unsigned long long)(uintptr_t)(W + (size_t)cb * FP);
    unsigned rows_left = (unsigned)(FP - cb);
    v4u g0;
    g0.x = 1u;                                    // count=1 (valid descriptor)
    g0.y = lds0;                                  // lds_addr
    g0.z = (unsigned)ga;                          // global_addr[31:0]
    g0.w = (unsigned)(ga >> 32) | (2u << 30);     // global_addr[56:32] | type=2
    v8i_t g1;
    g1[0] = (int)(2u << 16);                      // data_size = 2 (4 bytes)
    g1[1] = (int)((unsigned)FP << 16);            // tensor_dim0 = 200 (bits 63:48)
    g1[2] = (int)(rows_left << 16);               // tensor_dim1 (bits 111:80)
    g1[3] = (int)((unsigned)FP << 16);            // tile_dim0 = 200 (bits 127:112)
    g1[4] = 16;                                   // tile_dim1 = 16, tile_dim2 = 0
    g1[5] = FP;                                   // tensor_dim0_stride = 200
    g1[6] = 0;
    g1[7] = 0;
    v4i_t zz4 = {0, 0, 0, 0};
    v8i_t zz8 = {0, 0, 0, 0, 0, 0, 0, 0};
    __builtin_amdgcn_tensor_load_to_lds(g0, g1, zz4, zz4, zz8, 0);
    __builtin_amdgcn_s_wait_tensorcnt(0);
  }
  __syncthreads();

  int rb = (rg + (threadIdx.x >> 5)) << 4;
  int n = cb + lm;
  const float* xr = X + (size_t)(rb + lm) * FP;
  const float* wl = &sW[(size_t)lm * FP];
  v8f acc = {};
  for (int k0 = 0; k0 < FP; k0 += 4) {
    int ka = k0 + 2 * half;
    v2f a, b;
    a.x = xr[ka]; a.y = xr[ka + 1];
    b = *(const v2f*)&wl[ka];
    acc = __builtin_amdgcn_wmma_f32_16x16x4_f32(false, a, false, b, (short)0, acc, false, false);
  }
  if (n >= FP) return;
  float bv = bias[n];
#pragma unroll
  for (int v = 0; v < 8; ++v) {
    int m = rb + v + 8 * half;
    Y[(size_t)m * FP + n] = elu1(acc[v] + srow[m] * bv);
  }
}

// ---------------------------------------------------------------------------
// Kernel 4: fused GRU cell. Block = 4 row-tiles x 1 col-tile. The six weight
// sections (r/z/n of Wih and Whh, 16 rows x 200 each) are staged into LDS via
// async global->LDS copies (ASYNCcnt), rows padded to 204 dwords.
// Per wave: 6 WMMA accumulators, 300 v_wmma, pointwise gate epilogue.
// ---------------------------------------------------------------------------
__global__ void gru_kernel(const float* __restrict__ X,    // ctx [M,FP]
                           const float* __restrict__ H,    // h   [M,FP]
                           const float* __restrict__ Wih,  // [3FP,FP]
                           const float* __restrict__ Whh,  // [3FP,FP]
                           const float* __restrict__ bih,  // [3FP]
                           const float* __restrict__ bhh,  // [3FP]
                           float* __restrict__ Hn,         // [M,FP]
                           float* __restrict__ Act,        // [M,FP] or null
                           int M) {
  extern __shared__ float sB[];  // [6][16][WPAD] = 78336 B
  int cb = (blockIdx.x % NT_COLS) << 4;
  int rg = (blockIdx.x / NT_COLS) << 2;
  int lane = threadIdx.x & 31, half = lane >> 4, lm = lane & 15;

  // Stage 6 x (16 x 200 f32) weight tiles: 4800 x 16B chunks, async to LDS.
  const int CHUNKS = 6 * 16 * (FP / 4);  // 4800
  for (int c = threadIdx.x; c < CHUNKS; c += blockDim.x) {
    int s = c / 800, rem = c % 800, rr = rem / 50, q = rem % 50;
    const float* base = (s < 3) ? Wih : Whh;
    int sec = (s < 3) ? s : s - 3;
    int grow = cb + sec * FP + rr;
    if (grow > 3 * FP - 1) grow = 3 * FP - 1;   // clamp OOB rows (cols >=200 unused)
    unsigned long long gsrc = (unsigned long long)(uintptr_t)(base + (size_t)grow * FP + q * 4);
    unsigned ldst = (unsigned)(uintptr_t)&sB[((s * 16 + rr) * WPAD) + q * 4];
    asm volatile("global_load_async_to_lds_b128 %0, %1, off"
                 :: "v"(ldst), "v"(gsrc) : "memory");
  }
  asm volatile("s_wait_asynccnt 0x0" ::: "memory");
  __syncthreads();

  int rb = (rg + (threadIdx.x >> 5)) << 4;
  int n = cb + lm;
  const float* xr = X + (size_t)(rb + lm) * FP;
  const float* hr = H + (size_t)(rb + lm) * FP;
  const float* bir_p = &sB[(0 * 16 + lm) * WPAD];
  const float* biz_p = &sB[(1 * 16 + lm) * WPAD];
  const float* bin_p = &sB[(2 * 16 + lm) * WPAD];
  const float* bhr_p = &sB[(3 * 16 + lm) * WPAD];
  const float* bhz_p = &sB[(4 * 16 + lm) * WPAD];
  const float* bhn_p = &sB[(5 * 16 + lm) * WPAD];
  v8f air = {}, aiz = {}, ain = {}, ahr = {}, ahz = {}, ahn = {};
  for (int k0 = 0; k0 < FP; k0 += 4) {
    int ka = k0 + 2 * half;
    v2f ax, ah, b;
    ax.x = xr[ka]; ax.y = xr[ka + 1];
    ah.x = hr[ka]; ah.y = hr[ka + 1];
    b = *(const v2f*)&bir_p[ka];
    air = __builtin_amdgcn_wmma_f32_16x16x4_f32(false, ax, false, b, (short)0, air, false, false);
    b = *(const v2f*)&biz_p[ka];
    aiz = __builtin_amdgcn_wmma_f32_16x16x4_f32(false, ax, false, b, (short)0, aiz, false, false);
    b = *(const v2f*)&bin_p[ka];
    ain = __builtin_amdgcn_wmma_f32_16x16x4_f32(false, ax, false, b, (short)0, ain, false, false);
    b = *(const v2f*)&bhr_p[ka];
    ahr = __builtin_amdgcn_wmma_f32_16x16x4_f32(false, ah, false, b, (short)0, ahr, false, false);
    b = *(const v2f*)&bhz_p[ka];
    ahz = __builtin_amdgcn_wmma_f32_16x16x4_f32(false, ah, false, b, (short)0, ahz, false, false);
    b = *(const v2f*)&bhn_p[ka];
    ahn = __builtin_amdgcn_wmma_f32_16x16x4_f32(false, ah, false, b, (short)0, ahn, false, false);
  }
  if (n >= FP) return;
  float bir = bih[n], biz = bih[n + FP], bin = bih[n + 2 * FP];
  float bhr = bhh[n], bhz = bhh[n + FP], bhn = bhh[n + 2 * FP];
#pragma unroll
  for (int v = 0; v < 8; ++v) {
    int m = rb + v + 8 * half;
    float hv = H[(size_t)m * FP + n];
    float rg_ = sigmoidf(air[v] + bir + ahr[v] + bhr);
    float zg = sigmoidf(aiz[v] + biz + ahz[v] + bhz);
    float ng = tanhf(ain[v] + bin + rg_ * (ahn[v] + bhn));
    float o = (1.f - zg) * ng + zg * hv;
    Hn[(size_t)m * FP + n] = o;
    if (Act) Act[(size_t)m * FP + n] = fmaxf(o, 0.f);
  }
}

// ---------------------------------------------------------------------------
// Mol phase helpers
// ---------------------------------------------------------------------------
__global__ void mol_init_kernel(const float* __restrict__ act,
                                const float* __restrict__ amask,
                                float* __restrict__ mf,
                                float* __restrict__ act_mol) {
  int b = blockIdx.x;
  int k = threadIdx.x;
  if (k >= FP) return;
  float s = 0.f;
  for (int l = 0; l < LN; ++l)
    s += act[((size_t)b * LN + l) * FP + k] * amask[(size_t)b * LN + l];
  mf[(size_t)b * FP + k] = s;
  act_mol[(size_t)b * FP + k] = fmaxf(s, 0.f);
}

__global__ void mol_attn_kernel(const float* __restrict__ act,
                                const float* __restrict__ act_mol,
                                const float* __restrict__ amask,
                                const float* __restrict__ maw,      // [2*FP]
                                const float* __restrict__ mab,      // [1]
                                float* __restrict__ wact,
                                float* __restrict__ s_mw) {
  __shared__ float smw[LN];
  __shared__ float red[LN];
  int b = blockIdx.x;
  int l = threadIdx.x;  // 0..127
  const float* am = act_mol + (size_t)b * FP;
  const float* ar = act + ((size_t)b * LN + l) * FP;
  float p = mab[0];
  for (int k = 0; k < FP; ++k)
    p += am[k] * maw[k] + ar[k] * maw[FP + k];
  float msk = amask[(size_t)b * LN + l];
  float s = lrelu(p) + ((msk == 0.f) ? NEGBIG : 0.f);
  red[l] = s; __syncthreads();
  for (int off = LN >> 1; off > 0; off >>= 1) {
    if (l < off) red[l] = fmaxf(red[l], red[l + off]);
    __syncthreads();
  }
  float mx = red[0]; __syncthreads();
  float e = __expf(s - mx);
  red[l] = e; __syncthreads();
  for (int off = LN >> 1; off > 0; off >>= 1) {
    if (l < off) red[l] += red[l + off];
    __syncthreads();
  }
  float inv = 1.f / red[0]; __syncthreads();
  float mw = e * inv * msk;
  smw[l] = mw; red[l] = mw; __syncthreads();
  for (int off = LN >> 1; off > 0; off >>= 1) {
    if (l < off) red[l] += red[l + off];
    __syncthreads();
  }
  if (l == 0) s_mw[b] = red[0];
  __syncthreads();
  for (int k = l; k < FP; k += LN) {
    float acc = 0.f;
    for (int ll = 0; ll < LN; ++ll)
      acc += smw[ll] * act[((size_t)b * LN + ll) * FP + k];
    wact[(size_t)b * FP + k] = acc;
  }
}

__global__ void output_kernel(const float* __restrict__ mf,
                              const float* __restrict__ ow,
                              const float* __restrict__ ob,
                              float* __restrict__ out) {
  int b = blockIdx.x * blockDim.x + threadIdx.x;
  if (b >= BN) return;
  float s = ob[0];
  for (int k = 0; k < FP; ++k) s += mf[(size_t)b * FP + k] * ow[k];
  out[b] = s;
}

// ---------------------------------------------------------------------------
extern "C" void kernel_launch(void* const* d_in, const int* in_sizes, int n_in,
                              void* d_out, int out_size, void* d_ws, size_t ws_size,
                              hipStream_t stream) {
  (void)in_sizes; (void)n_in; (void)out_size; (void)ws_size;
  const float* atom_list     = (const float*)d_in[0];
  const float* bond_list     = (const float*)d_in[1];
  const int*   adl           = (const int*)d_in[2];
  const int*   bdl           = (const int*)d_in[3];
  const float* amask         = (const float*)d_in[4];
  const float* atom_fc_w     = (const float*)d_in[5];
  const float* atom_fc_b     = (const float*)d_in[6];
  const float* nfc_w         = (const float*)d_in[7];
  const float* nfc_b         = (const float*)d_in[8];
  const float* align_w       = (const float*)d_in[9];
  const float* align_b       = (const float*)d_in[10];
  const float* attend_w      = (const float*)d_in[11];
  const float* attend_b      = (const float*)d_in[12];
  const float* gru_wih       = (const float*)d_in[13];
  const float* gru_whh       = (const float*)d_in[14];
  const float* gru_bih       = (const float*)d_in[15];
  const float* gru_bhh       = (const float*)d_in[16];
  const float* mol_align_w   = (const float*)d_in[17];
  const float* mol_align_b   = (const float*)d_in[18];
  const float* mol_attend_w  = (const float*)d_in[19];
  const float* mol_attend_b  = (const float*)d_in[20];
  const float* mol_gru_wih   = (const float*)d_in[21];
  const float* mol_gru_whh   = (const float*)d_in[22];
  const float* mol_gru_bih   = (const float*)d_in[23];
  const float* mol_gru_bhh   = (const float*)d_in[24];
  const float* output_w      = (const float*)d_in[25];
  const float* output_b      = (const float*)d_in[26];

  float* out_atom = (float*)d_out;                       // [N,FP]
  float* out_pred = (float*)d_out + (size_t)NROWS * FP;  // [B]

  float* ws = (float*)d_ws;
  size_t o = 0;
  float* nbuf = ws + o; o += (size_t)NDROWS * FP;
  float* act  = ws + o; o += (size_t)NROWS * FP;
  float* wnb  = ws + o; o += (size_t)NROWS * FP;
  float* ctx  = ws + o; o += (size_t)NROWS * FP;
  float* hA   = ws + o; o += (size_t)NROWS * FP;
  float* hB   = ws + o; o += (size_t)NROWS * FP;
  float* s_aw = ws + o; o += (size_t)NROWS;
  float* mfA  = ws + o; o += (size_t)BN * FP;
  float* mfB  = ws + o; o += (size_t)BN * FP;
  float* actm = ws + o; o += (size_t)BN * FP;
  float* wact = ws + o; o += (size_t)BN * FP;
  float* s_mw = ws + o; o += (size_t)BN;
  float* mctx = ws + o; o += (size_t)BN * FP;

  const int blk = 128;  // 4 waves / block
  int gFC = ((NROWS  >> 4) * NT_COLS + 3) / 4;     // fc: wave-per-tile
  int gND = ((NDROWS >> 4) * NT_COLS + 3) / 4;
  int gN4 = (NROWS >> 6) * NT_COLS;                // block = 4 row-tiles x 1 col
  int gB4 = (BN    >> 6) * NT_COLS;
  const size_t shW   = 16 * FP * sizeof(float);        // 12800 B
  const size_t shGRU = 6 * 16 * WPAD * sizeof(float);  // 78336 B

  // 1) atom_fc -> h0
  fc_lrelu_kernel<<<gFC, blk, 0, stream>>>(atom_list, atom_fc_w, atom_fc_b, hA, NROWS, AFD);
  // 2) neighbor_fc -> nb buffer
  neighbor_fc_kernel<<<gND, blk, 0, stream>>>(atom_list, bond_list, adl, bdl, nfc_w, nfc_b, nbuf);

  // 3) radius loop
  float* h_cur = hA;
  for (int d = 0; d < 3; ++d) {
    float* h_next = (d == 0) ? hB : (d == 1) ? hA : out_atom;
    const float* score_cur = (d == 0) ? h_cur : act;
    const float* nbsrc     = (d == 0) ? nbuf : act;
    attn_score_kernel<<<NROWS / 8, 256, 0, stream>>>(
        score_cur, nbsrc, (d == 0) ? 0 : 1, adl,
        align_w + (size_t)d * 2 * FP, align_b + d, wnb, s_aw);
    attend_elu_kernel<<<gN4, blk, shW, stream>>>(
        wnb, attend_w + (size_t)d * FP * FP, attend_b + (size_t)d * FP, s_aw, ctx, NROWS);
    gru_kernel<<<gN4, blk, shGRU, stream>>>(
        ctx, h_cur,
        gru_wih + (size_t)d * 3 * FP * FP, gru_whh + (size_t)d * 3 * FP * FP,
        gru_bih + (size_t)d * 3 * FP,      gru_bhh + (size_t)d * 3 * FP,
        h_next, act, NROWS);
    h_cur = h_next;
  }

  // 4) mol phase
  mol_init_kernel<<<BN, 256, 0, stream>>>(act, amask, mfA, actm);
  float* mf_cur = mfA;
  for (int t = 0; t < 2; ++t) {
    float* mf_next = (t == 0) ? mfB : mfA;
    mol_attn_kernel<<<BN, LN, 0, stream>>>(act, actm, amask, mol_align_w, mol_align_b, wact, s_mw);
    attend_elu_kernel<<<gB4, blk, shW, stream>>>(wact, mol_attend_w, mol_attend_b, s_mw, mctx, BN);
    gru_kernel<<<gB4, blk, shGRU, stream>>>(mctx, mf_cur, mol_gru_wih, mol_gru_whh,
                                            mol_gru_bih, mol_gru_bhh, mf_next, actm, BN);
    mf_cur = mf_next;
  }
  output_kernel<<<1, 256, 0, stream>>>(mf_cur, output_w, output_b, out_pred);
}